// CrossSignalAnalyzer_17652315587313
// MI455X (gfx1250) — compile-verified
//
#include <hip/hip_runtime.h>
#include <hip/hip_bf16.h>
#include <math.h>

#define NSIG 2048
#define HID 16
#define BLOCK 256
#define NWAVE 8          // 256 / 32

typedef __attribute__((ext_vector_type(2))) float v2f;
typedef __attribute__((ext_vector_type(8))) float v8f;

__device__ __forceinline__ float gelu_erf(float x) {
    return 0.5f * x * (1.0f + erff(x * 0.70710678118654752440f));
}

// ---------------------------------------------------------------------------
// Kernel 1: per-signal scalars (hoists powf out of the O(n^2) loop).
// sig layout (SoA in d_ws): [field][NSIG], fields:
// 0:t(p5) 1:ra(p3) 2:dec(p4) 3:psi(p7) 4:chirp-mass 5:f_isco 6:dist
// ---------------------------------------------------------------------------
__global__ void precompute_sig(const float* __restrict__ p, float* __restrict__ sig) {
    int s = blockIdx.x * blockDim.x + threadIdx.x;
    if (s >= NSIG) return;
    const float* ps = p + s * 15;
    float m1 = ps[0] * 95.0f + 5.0f;
    float m2 = ps[1] * 95.0f + 5.0f;
    float mc = powf(m1 * m2, 0.6f) * powf(m1 + m2, -0.2f);
    float fisco = 220.0f / (m1 + m2);
    float dist = ps[2] * 2950.0f + 50.0f;
    sig[0 * NSIG + s] = ps[5];
    sig[1 * NSIG + s] = ps[3];
    sig[2 * NSIG + s] = ps[4];
    sig[3 * NSIG + s] = ps[7];
    sig[4 * NSIG + s] = mc;
    sig[5 * NSIG + s] = fisco;
    sig[6 * NSIG + s] = dist;
}

// ---------------------------------------------------------------------------
// Kernel 2: fully fused row kernel. One block per row i. 8 waves, each wave
// handles 256 j's in 8 iterations of 32 pairs (barrier-free inner loop):
//   - each lane computes the 8 pairwise features for its pair j
//   - B matrices built via intra-wave ds_bpermute shuffles (no LDS staging)
//   - two chained V_WMMA_F32_16X16X4_F32 per 16-pair group -> h[16hid x 16pr]
//   - LN(16) via 8-reg sum + shfl_xor(16), GELU, iw2 dot -> logit
//   - online softmax accumulation of (max, sum, sum e*feat[8]) per lane
// Then hierarchical (wave -> block) softmax-state merge, and the small
// overlap net (8->32->LN->GELU->16) finishes per-row.
// ---------------------------------------------------------------------------
__global__ void __launch_bounds__(BLOCK)
cross_signal_kernel(const float* __restrict__ sig,
                    const float* __restrict__ iw1, const float* __restrict__ ib1,
                    const float* __restrict__ ig1, const float* __restrict__ ibt1,
                    const float* __restrict__ iw2, const float* __restrict__ ib2,
                    const float* __restrict__ ow1, const float* __restrict__ ob1,
                    const float* __restrict__ og1, const float* __restrict__ obt1,
                    const float* __restrict__ ow2, const float* __restrict__ ob2,
                    float* __restrict__ out) {
    extern __shared__ float smem[];
    float* s_sig   = smem;                    // 7 * NSIG
    float* s_w1    = s_sig + 7 * NSIG;        // 128
    float* s_ib1   = s_w1 + 128;              // 16
    float* s_ig1   = s_ib1 + 16;              // 16
    float* s_ibt1  = s_ig1 + 16;              // 16
    float* s_iw2   = s_ibt1 + 16;             // 16
    float* s_red   = s_iw2 + 16;              // NWAVE * 12
    float* s_g2    = s_red + NWAVE * 12;      // 32
    float* s_wtd   = s_g2 + 32;               // 8

    const int tid = threadIdx.x;
    const int i = blockIdx.x;
    const int wv = tid >> 5;
    const int L = tid & 31;
    const int m16 = L & 15;
    const bool lowHalf = (L < 16);
    const int hibase = (L >> 4) * 8;   // hid rows this lane-half holds in D
    const int kh = (L >> 4) * 2;       // K slot within a K=4 chunk

    // ---- load per-signal scalars + importance-net weights into LDS ----
    for (int k = tid; k < 7 * NSIG; k += BLOCK) s_sig[k] = sig[k];
    if (tid < 128) s_w1[tid] = iw1[tid];
    if (tid < 16) {
        s_ib1[tid] = ib1[tid];
        s_ig1[tid] = ig1[tid];
        s_ibt1[tid] = ibt1[tid];
        s_iw2[tid] = iw2[tid];
    }
    __syncthreads();

    // row-i scalars
    const float ti   = s_sig[0 * NSIG + i];
    const float rai  = s_sig[1 * NSIG + i];
    const float deci = s_sig[2 * NSIG + i];
    const float psii = s_sig[3 * NSIG + i];
    const float mci  = s_sig[4 * NSIG + i];
    const float fii  = s_sig[5 * NSIG + i];
    const float di   = s_sig[6 * NSIG + i];
    const float ib2v = ib2[0];

    // ---- A matrix (iw1, 16x8) in WMMA A layout, hoisted: two K=4 chunks ----
    // A 16x4 f32 layout: lane = M (0-15 twice); VGPR0 holds K = kh, VGPR1 K = kh+1
    v2f A0, A1;
    A0[0] = s_w1[m16 * 8 + kh];
    A0[1] = s_w1[m16 * 8 + kh + 1];
    A1[0] = s_w1[m16 * 8 + 4 + kh];
    A1[1] = s_w1[m16 * 8 + 4 + kh + 1];

    // online softmax state (this lane owns pair j = wv*256 + it*32 + L)
    float mrun = -INFINITY, srun = 0.0f;
    float wacc[8] = {0, 0, 0, 0, 0, 0, 0, 0};

    for (int it = 0; it < 8; ++it) {
        const int j = wv * 256 + it * 32 + L;

        // ---- pairwise features for (i, j) ----
        const float tj   = s_sig[0 * NSIG + j];
        const float raj  = s_sig[1 * NSIG + j];
        const float decj = s_sig[2 * NSIG + j];
        const float psij = s_sig[3 * NSIG + j];
        const float mcj  = s_sig[4 * NSIG + j];
        const float fij  = s_sig[5 * NSIG + j];
        const float dj   = s_sig[6 * NSIG + j];

        float ft[8];
        const float dra = fabsf(rai - raj);
        const float ddec = fabsf(deci - decj);
        ft[0] = fabsf(ti - tj);
        ft[1] = sqrtf(dra * dra + ddec * ddec);
        ft[2] = 1.0f / (1.0f + fabsf(mci - mcj) * (1.0f / 30.0f));
        ft[3] = expf(-fabsf(fii - fij) * 0.01f);
        ft[4] = fminf(di, dj) / fmaxf(di, dj);
        ft[5] = fabsf(psii - psij);
        ft[6] = dra;
        ft[7] = ddec;

        // ---- two 16-pair groups through WMMA f32 16x16x4 (K=8 in 2 chunks) ----
        // B gather is intra-wave: lane L needs feats[pair g*16+m16][f] where
        // f = kh+r -> two bpermutes + select per B register (no LDS, no barrier).
        float logit2[2];
#pragma unroll
        for (int g = 0; g < 2; ++g) {
            const int src = g * 16 + m16;
            v2f B0, B1;
            {
                const float x0 = __shfl(ft[0], src, 32);
                const float x2 = __shfl(ft[2], src, 32);
                B0[0] = lowHalf ? x0 : x2;                 // f = kh
                const float x1 = __shfl(ft[1], src, 32);
                const float x3 = __shfl(ft[3], src, 32);
                B0[1] = lowHalf ? x1 : x3;                 // f = kh+1
                const float x4 = __shfl(ft[4], src, 32);
                const float x6 = __shfl(ft[6], src, 32);
                B1[0] = lowHalf ? x4 : x6;                 // f = 4+kh
                const float x5 = __shfl(ft[5], src, 32);
                const float x7 = __shfl(ft[7], src, 32);
                B1[1] = lowHalf ? x5 : x7;                 // f = 5+kh
            }

            v8f acc = {};
            acc = __builtin_amdgcn_wmma_f32_16x16x4_f32(
                false, A0, false, B0, (short)0, acc, false, false);
            acc = __builtin_amdgcn_wmma_f32_16x16x4_f32(
                false, A1, false, B1, (short)0, acc, false, false);

            // LayerNorm over 16 hidden: 8 regs here + partner lane (xor 16)
            float v[8], sum = 0.0f, sum2 = 0.0f;
#pragma unroll
            for (int r = 0; r < 8; ++r) {
                v[r] = acc[r] + s_ib1[r + hibase];
                sum += v[r];
                sum2 += v[r] * v[r];
            }
            sum += __shfl_xor(sum, 16, 32);
            sum2 += __shfl_xor(sum2, 16, 32);
            const float mean = sum * (1.0f / 16.0f);
            const float var = sum2 * (1.0f / 16.0f) - mean * mean;
            const float inv = rsqrtf(var + 1e-5f);

            float part = 0.0f;
#pragma unroll
            for (int r = 0; r < 8; ++r) {
                const int h = r + hibase;
                const float x = (v[r] - mean) * inv * s_ig1[h] + s_ibt1[h];
                part += s_iw2[h] * gelu_erf(x);
            }
            part += __shfl_xor(part, 16, 32);
            logit2[g] = part + ib2v;
        }

        // lane L owns pair jbase+L: group0 for L<16, group1 for L>=16
        const float logit = lowHalf ? logit2[0] : logit2[1];

        // ---- online softmax update (skip self-pair) ----
        if (j != i) {
            const float nm = fmaxf(mrun, logit);
            const float sc = expf(mrun - nm);   // 0 when mrun == -inf
            const float e = expf(logit - nm);
            srun = srun * sc + e;
#pragma unroll
            for (int f = 0; f < 8; ++f) wacc[f] = wacc[f] * sc + e * ft[f];
            mrun = nm;
        }
    }

    // ---- wave-level softmax-state merge ----
#pragma unroll
    for (int off = 16; off; off >>= 1) {
        const float mo = __shfl_xor(mrun, off, 32);
        const float so = __shfl_xor(srun, off, 32);
        const float nm = fmaxf(mrun, mo);
        const float e1 = expf(mrun - nm);
        const float e2 = expf(mo - nm);
        srun = srun * e1 + so * e2;
#pragma unroll
        for (int f = 0; f < 8; ++f) {
            const float wo = __shfl_xor(wacc[f], off, 32);
            wacc[f] = wacc[f] * e1 + wo * e2;
        }
        mrun = nm;
    }
    if (L == 0) {
        float* rp = &s_red[wv * 12];
        rp[0] = mrun;
        rp[1] = srun;
#pragma unroll
        for (int f = 0; f < 8; ++f) rp[2 + f] = wacc[f];
    }
    __syncthreads();

    // ---- block-level merge + normalize ----
    if (tid == 0) {
        float M = s_red[0], S = s_red[1], W[8];
#pragma unroll
        for (int f = 0; f < 8; ++f) W[f] = s_red[2 + f];
        for (int w2 = 1; w2 < NWAVE; ++w2) {
            const float* rp = &s_red[w2 * 12];
            const float nm = fmaxf(M, rp[0]);
            const float e1 = expf(M - nm);
            const float e2 = expf(rp[0] - nm);
            S = S * e1 + rp[1] * e2;
#pragma unroll
            for (int f = 0; f < 8; ++f) W[f] = W[f] * e1 + rp[2 + f] * e2;
            M = nm;
        }
        const float invS = 1.0f / S;
#pragma unroll
        for (int f = 0; f < 8; ++f) s_wtd[f] = W[f] * invS;
    }
    __syncthreads();

    // ---- overlap net: 8 -> 32 -> LN -> GELU (wave 0) ----
    if (tid < 32) {
        float h = ob1[tid];
#pragma unroll
        for (int f = 0; f < 8; ++f) h += s_wtd[f] * ow1[tid * 8 + f];
        float sum = h, sum2 = h * h;
#pragma unroll
        for (int off = 16; off; off >>= 1) {
            sum += __shfl_xor(sum, off, 32);
            sum2 += __shfl_xor(sum2, off, 32);
        }
        const float mean = sum * (1.0f / 32.0f);
        const float var = sum2 * (1.0f / 32.0f) - mean * mean;
        const float x = (h - mean) * rsqrtf(var + 1e-5f) * og1[tid] + obt1[tid];
        s_g2[tid] = gelu_erf(x);
    }
    __syncthreads();

    // ---- final 32 -> 16 ----
    if (tid < 16) {
        float o = ob2[tid];
#pragma unroll
        for (int k = 0; k < 32; ++k) o += ow2[tid * 32 + k] * s_g2[k];
        out[i * 16 + tid] = o;
    }
}

extern "C" void kernel_launch(void* const* d_in, const int* in_sizes, int n_in,
                              void* d_out, int out_size, void* d_ws, size_t ws_size,
                              hipStream_t stream) {
    (void)in_sizes; (void)n_in; (void)out_size; (void)ws_size;
    const float* params = (const float*)d_in[0];
    const float* iw1  = (const float*)d_in[1];
    const float* ib1  = (const float*)d_in[2];
    const float* ig1  = (const float*)d_in[3];
    const float* ibt1 = (const float*)d_in[4];
    const float* iw2  = (const float*)d_in[5];
    const float* ib2  = (const float*)d_in[6];
    const float* ow1  = (const float*)d_in[7];
    const float* ob1  = (const float*)d_in[8];
    const float* og1  = (const float*)d_in[9];
    const float* obt1 = (const float*)d_in[10];
    const float* ow2  = (const float*)d_in[11];
    const float* ob2  = (const float*)d_in[12];
    float* out = (float*)d_out;
    float* sig = (float*)d_ws;   // 7 * NSIG floats = 57 KB

    precompute_sig<<<(NSIG + BLOCK - 1) / BLOCK, BLOCK, 0, stream>>>(params, sig);

    const size_t lds_bytes =
        (7 * NSIG + 128 + 16 * 4 + NWAVE * 12 + 32 + 8) * sizeof(float); // ~58.7 KB
    cross_signal_kernel<<<NSIG, BLOCK, lds_bytes, stream>>>(
        sig, iw1, ib1, ig1, ibt1, iw2, ib2,
        ow1, ob1, og1, obt1, ow2, ob2, out);
}